// SS2D_39290360823833
// MI455X (gfx1250) — compile-verified
//
#include <hip/hip_runtime.h>
#include <math.h>

// ---------------------------------------------------------------------------
// SS2D (VMamba selective-scan 2D) forward for gfx1250 (MI455X, wave32, WMMA).
// Shapes: B=2, C=96, H=W=32, L=1024, Din=192, N=16, R=6, K=4 directions.
// GEMMs: V_WMMA_F32_16X16X4_F32 (fp32 matrix core, confirmed lowering).
// Scan: 3-phase chunked parallel linear recurrence (32 chunks x 32 steps).
// ---------------------------------------------------------------------------

typedef float v2f __attribute__((ext_vector_type(2)));
typedef float v8f __attribute__((ext_vector_type(8)));

#define BSZ   2
#define CMOD  96
#define LSEQ  1024
#define DIN   192
#define NST   16
#define RRK   6
#define KDIR  4
#define NCH   32      // chunks
#define CHL   32      // chunk length (NCH*CHL == LSEQ)

__device__ __forceinline__ int tmap(int l) { return ((l & 31) << 5) | (l >> 5); } // H=W=32 transpose (involution)
__device__ __forceinline__ int dirmap(int k, int l) {
    // xs[:,0]=x_hw, xs[:,1]=x_wh, xs[:,2]=rev(x_hw), xs[:,3]=rev(x_wh)
    switch (k) {
        case 0:  return l;
        case 1:  return tmap(l);
        case 2:  return (LSEQ - 1) - l;
        default: return tmap((LSEQ - 1) - l);
    }
}

// ============================================================================
// K1: in_proj GEMM.  xi[m=(b,l)][d] = sum_c x[b,c,l] * in_proj_w[d][c]
//     M=2048, N=192 (first Din rows of in_proj_w), Kd=96.  One wave per tile.
// ============================================================================
__global__ void k_inproj(const float* __restrict__ x, const float* __restrict__ w,
                         float* __restrict__ xi) {
    const int lane  = threadIdx.x;       // 0..31
    const int row   = lane & 15;
    const int khalf = lane >> 4;         // 0/1
    const int m0 = blockIdx.x * 16;      // over BL=2048
    const int n0 = blockIdx.y * 16;      // over Din=192
    const int m  = m0 + row;
    const int b  = m >> 10;
    const int l  = m & 1023;

    v8f acc = {};
    for (int kb = 0; kb < CMOD; kb += 4) {
        const int k0 = kb + 2 * khalf;
        v2f a, bb;
        a.x  = x[((size_t)b * CMOD + (k0 + 0)) * LSEQ + l];
        a.y  = x[((size_t)b * CMOD + (k0 + 1)) * LSEQ + l];
        bb.x = w[(size_t)(n0 + row) * CMOD + (k0 + 0)];
        bb.y = w[(size_t)(n0 + row) * CMOD + (k0 + 1)];
        acc = __builtin_amdgcn_wmma_f32_16x16x4_f32(false, a, false, bb, (short)0, acc, false, false);
    }
#pragma unroll
    for (int v = 0; v < 8; ++v) {
        const int mm = m0 + v + 8 * khalf;
        xi[(size_t)mm * DIN + (n0 + row)] = acc[v];
    }
}

// ============================================================================
// K2: depthwise 3x3 conv (SAME, zero pad) + bias + SiLU.  (b,l,d) layout.
// ============================================================================
__global__ void k_conv_silu(const float* __restrict__ xi, const float* __restrict__ cw,
                            const float* __restrict__ cb, float* __restrict__ xa) {
    const int idx = blockIdx.x * blockDim.x + threadIdx.x;
    if (idx >= BSZ * LSEQ * DIN) return;
    const int d = idx % DIN;
    const int l = (idx / DIN) % LSEQ;
    const int b = idx / (DIN * LSEQ);
    const int h = l >> 5, w = l & 31;

    float s = cb[d];
#pragma unroll
    for (int i = 0; i < 3; ++i) {
#pragma unroll
        for (int j = 0; j < 3; ++j) {
            const int hh = h + i - 1, ww = w + j - 1;
            if (hh >= 0 && hh < 32 && ww >= 0 && ww < 32)
                s += cw[d * 9 + i * 3 + j] *
                     xi[((size_t)b * LSEQ + (hh * 32 + ww)) * DIN + d];
        }
    }
    xa[idx] = s / (1.0f + __expf(-s));   // SiLU
}

// ============================================================================
// K3: x_dbl = einsum('bkdl,kdr->bkrl').  Per direction k:
//     M=2048 (b,l), N=38 (pad to 48), Kd=192.  A gathered from xa via dirmap.
//     Output layout xdbl[k][b][r][l], r in [0,38): 0..5 dts, 6..21 Bs, 22..37 Cs.
// ============================================================================
__global__ void k_xdbl(const float* __restrict__ xa, const float* __restrict__ xpw,
                       float* __restrict__ xdbl) {
    const int lane  = threadIdx.x;
    const int row   = lane & 15;
    const int khalf = lane >> 4;
    const int m0 = blockIdx.x * 16;
    const int n0 = blockIdx.y * 16;       // 0,16,32 (N padded 38->48)
    const int kd = blockIdx.z;            // direction
    const int m  = m0 + row;
    const int b  = m >> 10;
    const int l  = m & 1023;
    const int lsrc = dirmap(kd, l);
    const int n  = n0 + row;

    v8f acc = {};
    for (int kb = 0; kb < DIN; kb += 4) {
        const int k0 = kb + 2 * khalf;
        v2f a, bb;
        a.x  = xa[((size_t)b * LSEQ + lsrc) * DIN + (k0 + 0)];
        a.y  = xa[((size_t)b * LSEQ + lsrc) * DIN + (k0 + 1)];
        bb.x = (n < 38) ? xpw[((size_t)kd * DIN + (k0 + 0)) * 38 + n] : 0.0f;
        bb.y = (n < 38) ? xpw[((size_t)kd * DIN + (k0 + 1)) * 38 + n] : 0.0f;
        acc = __builtin_amdgcn_wmma_f32_16x16x4_f32(false, a, false, bb, (short)0, acc, false, false);
    }
    if (n < 38) {
#pragma unroll
        for (int v = 0; v < 8; ++v) {
            const int mm = m0 + v + 8 * khalf;
            const int b2 = mm >> 10, l2 = mm & 1023;
            xdbl[(((size_t)kd * BSZ + b2) * 38 + n) * LSEQ + l2] = acc[v];
        }
    }
}

// ============================================================================
// K4: dts = einsum('bkrl,kdr->bkdl') + 2*dt_projs_b, then softplus -> delta.
//     Per direction: M=2048, N=192, Kd=6 (pad to 8).  delta layout [b][k][l][d].
// ============================================================================
__global__ void k_dtproj(const float* __restrict__ xdbl, const float* __restrict__ dtw,
                         const float* __restrict__ dtb, float* __restrict__ delta) {
    const int lane  = threadIdx.x;
    const int row   = lane & 15;
    const int khalf = lane >> 4;
    const int m0 = blockIdx.x * 16;
    const int n0 = blockIdx.y * 16;
    const int kd = blockIdx.z;
    const int m  = m0 + row;
    const int b  = m >> 10;
    const int l  = m & 1023;
    const int n  = n0 + row;

    v8f acc = {};
    for (int kb = 0; kb < 8; kb += 4) {
        const int k0 = kb + 2 * khalf;
        v2f a, bb;
        a.x  = (k0 + 0 < RRK) ? xdbl[(((size_t)kd * BSZ + b) * 38 + (k0 + 0)) * LSEQ + l] : 0.0f;
        a.y  = (k0 + 1 < RRK) ? xdbl[(((size_t)kd * BSZ + b) * 38 + (k0 + 1)) * LSEQ + l] : 0.0f;
        bb.x = (k0 + 0 < RRK) ? dtw[((size_t)kd * DIN + n) * RRK + (k0 + 0)] : 0.0f;
        bb.y = (k0 + 1 < RRK) ? dtw[((size_t)kd * DIN + n) * RRK + (k0 + 1)] : 0.0f;
        acc = __builtin_amdgcn_wmma_f32_16x16x4_f32(false, a, false, bb, (short)0, acc, false, false);
    }
    const float bias2 = 2.0f * dtb[kd * DIN + n];   // bias applied twice in reference
#pragma unroll
    for (int v = 0; v < 8; ++v) {
        const int mm = m0 + v + 8 * khalf;
        const int b2 = mm >> 10, l2 = mm & 1023;
        const float xv = acc[v] + bias2;
        const float sp = (xv > 20.0f) ? xv : log1pf(__expf(xv));   // softplus
        delta[(((size_t)b2 * KDIR + kd) * LSEQ + l2) * DIN + n] = sp;
    }
}

// ============================================================================
// Selective scan, 3-phase chunked parallel linear recurrence.
// Recurrence per (b,k,d,n):  h_l = dA_l * h_{l-1} + du_l * B_l[n].
// Phase 1: per 32-step chunk compute affine composition (P = prod dA, S = run
//          from h=0).  256 blocks x 192 threads (vs 8 blocks sequential).
// Phase 2: serial prefix over 32 chunks -> incoming state h0 per chunk.
// Phase 3: re-run each chunk from exact h0, emit y (identical in-chunk
//          arithmetic ordering to the sequential reference).
// ============================================================================
__global__ void k_scan_chunk(const float* __restrict__ xa, const float* __restrict__ xdbl,
                             const float* __restrict__ delta, const float* __restrict__ A_logs,
                             float* __restrict__ Pbuf, float* __restrict__ Sbuf) {
    const int c  = blockIdx.x;          // chunk 0..31
    const int bk = blockIdx.y;          // 0..7
    const int b  = bk >> 2, kd = bk & 3;
    const int d  = threadIdx.x;         // 0..191
    const int l0 = c * CHL;

    __shared__ float sB[CHL * NST];     // B_l[n] for the whole chunk
    const float* xdk = xdbl + ((size_t)kd * BSZ + b) * 38 * LSEQ;
    for (int i = threadIdx.x; i < CHL * NST; i += DIN) {
        const int n = i >> 5, s = i & 31;          // n-major -> contiguous global reads
        sB[s * NST + n] = xdk[(size_t)(RRK + n) * LSEQ + l0 + s];
    }
    __syncthreads();

    float An[NST], P[NST], h[NST];
#pragma unroll
    for (int n = 0; n < NST; ++n) {
        An[n] = -__expf(A_logs[((size_t)kd * DIN + d) * NST + n]);
        P[n]  = 1.0f;
        h[n]  = 0.0f;
    }
    for (int s = 0; s < CHL; ++s) {
        const int l    = l0 + s;
        const int lsrc = dirmap(kd, l);
        const float u  = xa[((size_t)b * LSEQ + lsrc) * DIN + d];
        const float dl = delta[(((size_t)b * KDIR + kd) * LSEQ + l) * DIN + d];
        const float du = dl * u;
#pragma unroll
        for (int n = 0; n < NST; ++n) {
            const float dA = __expf(dl * An[n]);
            P[n] *= dA;
            h[n]  = h[n] * dA + du * sB[s * NST + n];
        }
    }
    const size_t base = (((size_t)bk * NCH + c) * DIN + d) * NST;
#pragma unroll
    for (int n = 0; n < NST; ++n) { Pbuf[base + n] = P[n]; Sbuf[base + n] = h[n]; }
}

__global__ void k_scan_prefix(const float* __restrict__ Pbuf, const float* __restrict__ Sbuf,
                              float* __restrict__ H0) {
    const int idx = blockIdx.x * blockDim.x + threadIdx.x;   // over 8*192*16 = 24576
    if (idx >= BSZ * KDIR * DIN * NST) return;
    const int bk  = idx / (DIN * NST);
    const int rem = idx % (DIN * NST);
    float carry = 0.0f;
    for (int c = 0; c < NCH; ++c) {
        const size_t i = (size_t)(bk * NCH + c) * (DIN * NST) + rem;
        H0[i] = carry;
        carry = carry * Pbuf[i] + Sbuf[i];
    }
}

__global__ void k_scan_apply(const float* __restrict__ xa, const float* __restrict__ xdbl,
                             const float* __restrict__ delta, const float* __restrict__ A_logs,
                             const float* __restrict__ Ds, const float* __restrict__ H0,
                             float* __restrict__ outy) {
    const int c  = blockIdx.x;
    const int bk = blockIdx.y;
    const int b  = bk >> 2, kd = bk & 3;
    const int d  = threadIdx.x;
    const int l0 = c * CHL;

    __shared__ float sB[CHL * NST];
    __shared__ float sC[CHL * NST];
    const float* xdk = xdbl + ((size_t)kd * BSZ + b) * 38 * LSEQ;
    for (int i = threadIdx.x; i < CHL * NST; i += DIN) {
        const int n = i >> 5, s = i & 31;
        sB[s * NST + n] = xdk[(size_t)(RRK + n) * LSEQ + l0 + s];
        sC[s * NST + n] = xdk[(size_t)(RRK + NST + n) * LSEQ + l0 + s];
    }
    __syncthreads();

    const size_t base = (((size_t)bk * NCH + c) * DIN + d) * NST;
    float An[NST], h[NST];
#pragma unroll
    for (int n = 0; n < NST; ++n) {
        An[n] = -__expf(A_logs[((size_t)kd * DIN + d) * NST + n]);
        h[n]  = H0[base + n];
    }
    const float Dv = Ds[kd * DIN + d];

    for (int s = 0; s < CHL; ++s) {
        const int l    = l0 + s;
        const int lsrc = dirmap(kd, l);
        const float u  = xa[((size_t)b * LSEQ + lsrc) * DIN + d];
        const float dl = delta[(((size_t)b * KDIR + kd) * LSEQ + l) * DIN + d];
        const float du = dl * u;
        float y = 0.0f;
#pragma unroll
        for (int n = 0; n < NST; ++n) {
            const float dA = __expf(dl * An[n]);
            h[n] = h[n] * dA + du * sB[s * NST + n];
            y += h[n] * sC[s * NST + n];
        }
        outy[(((size_t)b * KDIR + kd) * LSEQ + l) * DIN + d] = y + Dv * u;
    }
}

// ============================================================================
// K6: merge 4 directions (+reversal / H-W transpose), /4, LayerNorm over Din.
//     One block per (b,p), 192 threads. Deterministic serial reductions.
// ============================================================================
__global__ void k_merge_ln(const float* __restrict__ outy, const float* __restrict__ gamma,
                           const float* __restrict__ beta, float* __restrict__ yn) {
    const int bp = blockIdx.x;          // 0..2047
    const int b  = bp >> 10, p = bp & 1023;
    const int d  = threadIdx.x;
    const int tp = tmap(p);

    const float g0 = outy[(((size_t)b * KDIR + 0) * LSEQ + p) * DIN + d];
    const float g1 = outy[(((size_t)b * KDIR + 2) * LSEQ + (LSEQ - 1 - p)) * DIN + d];
    const float g2 = outy[(((size_t)b * KDIR + 1) * LSEQ + tp) * DIN + d];
    const float g3 = outy[(((size_t)b * KDIR + 3) * LSEQ + (LSEQ - 1 - tp)) * DIN + d];
    const float yv = 0.25f * (g0 + g1 + g2 + g3);

    __shared__ float sh[DIN];
    __shared__ float smu, svar;
    sh[d] = yv;
    __syncthreads();
    if (d == 0) { float s = 0.f; for (int i = 0; i < DIN; ++i) s += sh[i]; smu = s / DIN; }
    __syncthreads();
    const float dif = yv - smu;
    sh[d] = dif * dif;
    __syncthreads();
    if (d == 0) { float s = 0.f; for (int i = 0; i < DIN; ++i) s += sh[i]; svar = s / DIN; }
    __syncthreads();
    yn[(size_t)bp * DIN + d] = dif * rsqrtf(svar + 1e-5f) * gamma[d] + beta[d];
}

// ============================================================================
// K7: out_proj GEMM.  out[b,c,l] = sum_d yn[(b,l)][d] * out_proj_w[c][d]
//     M=2048, N=96, Kd=192.
// ============================================================================
__global__ void k_outproj(const float* __restrict__ yn, const float* __restrict__ w,
                          float* __restrict__ out) {
    const int lane  = threadIdx.x;
    const int row   = lane & 15;
    const int khalf = lane >> 4;
    const int m0 = blockIdx.x * 16;
    const int n0 = blockIdx.y * 16;   // over C=96
    const int m  = m0 + row;

    v8f acc = {};
    for (int kb = 0; kb < DIN; kb += 4) {
        const int k0 = kb + 2 * khalf;
        v2f a, bb;
        a.x  = yn[(size_t)m * DIN + (k0 + 0)];
        a.y  = yn[(size_t)m * DIN + (k0 + 1)];
        bb.x = w[(size_t)(n0 + row) * DIN + (k0 + 0)];
        bb.y = w[(size_t)(n0 + row) * DIN + (k0 + 1)];
        acc = __builtin_amdgcn_wmma_f32_16x16x4_f32(false, a, false, bb, (short)0, acc, false, false);
    }
#pragma unroll
    for (int v = 0; v < 8; ++v) {
        const int mm = m0 + v + 8 * khalf;
        const int b2 = mm >> 10, l2 = mm & 1023;
        out[((size_t)b2 * CMOD + (n0 + row)) * LSEQ + l2] = acc[v];
    }
}

// ============================================================================
// Launcher
// ============================================================================
extern "C" void kernel_launch(void* const* d_in, const int* in_sizes, int n_in,
                              void* d_out, int out_size, void* d_ws, size_t ws_size,
                              hipStream_t stream) {
    const float* x       = (const float*)d_in[0];
    const float* inw     = (const float*)d_in[1];
    const float* convw   = (const float*)d_in[2];
    const float* convb   = (const float*)d_in[3];
    const float* xpw     = (const float*)d_in[4];
    const float* dtw     = (const float*)d_in[5];
    const float* dtb     = (const float*)d_in[6];
    const float* A_logs  = (const float*)d_in[7];
    const float* Ds      = (const float*)d_in[8];
    const float* gamma   = (const float*)d_in[9];
    const float* beta    = (const float*)d_in[10];
    const float* outw    = (const float*)d_in[11];
    float* out = (float*)d_out;

    // Workspace partition (floats); total ~27 MB, L2-resident on MI455X.
    float* ws    = (float*)d_ws;
    float* xi    = ws;                       // 2*1024*192      =  393216
    float* xa    = xi    + 393216;           //                   393216
    float* xdbl  = xa    + 393216;           // 4*2*38*1024     =  311296
    float* delta = xdbl  + 311296;           // 2*4*1024*192    = 1572864
    float* outy  = delta + 1572864;          //                  1572864
    float* yn    = outy  + 1572864;          //                   393216
    float* Pbuf  = yn    + 393216;           // 8*32*192*16     =  786432
    float* Sbuf  = Pbuf  + 786432;           //                   786432
    float* H0    = Sbuf  + 786432;           //                   786432

    // 1) in_proj GEMM (WMMA f32)
    k_inproj<<<dim3(128, 12), 32, 0, stream>>>(x, inw, xi);

    // 2) depthwise conv + SiLU
    k_conv_silu<<<(BSZ * LSEQ * DIN + 255) / 256, 256, 0, stream>>>(xi, convw, convb, xa);

    // 3) x_dbl projection per direction (WMMA f32)
    k_xdbl<<<dim3(128, 3, 4), 32, 0, stream>>>(xa, xpw, xdbl);

    // 4) dt projection + softplus (WMMA f32)
    k_dtproj<<<dim3(128, 12, 4), 32, 0, stream>>>(xdbl, dtw, dtb, delta);

    // 5) selective scan: chunked parallel linear recurrence
    k_scan_chunk <<<dim3(NCH, BSZ * KDIR), DIN, 0, stream>>>(xa, xdbl, delta, A_logs, Pbuf, Sbuf);
    k_scan_prefix<<<(BSZ * KDIR * DIN * NST + 255) / 256, 256, 0, stream>>>(Pbuf, Sbuf, H0);
    k_scan_apply <<<dim3(NCH, BSZ * KDIR), DIN, 0, stream>>>(xa, xdbl, delta, A_logs, Ds, H0, outy);

    // 6) direction merge + LayerNorm
    k_merge_ln<<<BSZ * LSEQ, DIN, 0, stream>>>(outy, gamma, beta, yn);

    // 7) out_proj GEMM (WMMA f32)
    k_outproj<<<dim3(128, 6), 32, 0, stream>>>(yn, outw, out);
}